// KeGATv2_72112500900245
// MI455X (gfx1250) — compile-verified
//
#include <hip/hip_runtime.h>
#include <math.h>

// GATv2 (2 layers) + KnowledgeEnhancer for MI455X (gfx1250, wave32).
// Dense projections via V_WMMA_F32_16X16X4_F32; edge softmax/aggregation via
// native f32 global atomics. Softmax max-shift omitted (self-loops guarantee
// a non-empty segment; shift cancels algebraically).

typedef float v2f __attribute__((ext_vector_type(2)));
typedef float v8f __attribute__((ext_vector_type(8)));

#define NEG_SLOPE 0.2f

// ---------------------------------------------------------------------------
// WMMA GEMM: C[M,Ncols] = act(A[M,K] @ B[K,Ncols] (+bias)) (+skip)
// ACT: 0 = none/no bias, 1 = relu(+bias), 2 = skip + sigmoid(+bias)
// One wave per 16x16 output tile. Requires K % 4 == 0, Ncols % 16 == 0.
// VGPR layouts per CDNA5 ISA 7.12.2:
//   A: lane L -> M = L&15, vgpr i -> K = k0 + 2*(L>>4) + i
//   B: lane L -> N = L&15, vgpr i -> K = k0 + 2*(L>>4) + i
//   C/D: lane L -> N = L&15, vgpr r -> M = r + 8*(L>>4)
// ---------------------------------------------------------------------------
template<int ACT>
__global__ __launch_bounds__(32)
void gemm_wmma(const float* __restrict__ A, const float* __restrict__ B,
               const float* __restrict__ bias, const float* __restrict__ skip,
               float* __restrict__ C, int M, int K, int Ncols)
{
  const int lane = threadIdx.x;        // 0..31
  const int half = lane >> 4;          // 0 or 1
  const int l16  = lane & 15;
  const int colBase = blockIdx.x * 16;
  const int rowBase = blockIdx.y * 16;

  int arow = rowBase + l16;
  if (arow >= M) arow = M - 1;         // clamp (results discarded on store)
  const float* Aptr = A + (size_t)arow * K + 2 * half;
  const float* Bptr = B + (size_t)(2 * half) * Ncols + colBase + l16;

  v8f acc = {};
  for (int k = 0; k < K; k += 4) {
    v2f a = *reinterpret_cast<const v2f*>(Aptr + k);   // K = k+2*half, k+2*half+1
    v2f b;
    b.x = Bptr[(size_t)k * Ncols];
    b.y = Bptr[(size_t)(k + 1) * Ncols];
    acc = __builtin_amdgcn_wmma_f32_16x16x4_f32(false, a, false, b,
                                                (short)0, acc, false, false);
  }

  const int n = colBase + l16;
  float bv = 0.0f;
  if (ACT != 0) bv = bias[n];
  #pragma unroll
  for (int r = 0; r < 8; ++r) {
    const int m = rowBase + r + 8 * half;
    if (m < M) {
      float v = acc[r] + bv;
      if (ACT == 1) v = v > 0.0f ? v : 0.0f;
      if (ACT == 2) {
        const float g = 1.0f / (1.0f + expf(-v));
        v = skip[(size_t)m * Ncols + n] + g;
      }
      C[(size_t)m * Ncols + n] = v;
    }
  }
}

__device__ __forceinline__ float waveSum(float v) {
  #pragma unroll
  for (int off = 16; off >= 1; off >>= 1) v += __shfl_xor(v, off, 32);
  return v;
}
__device__ __forceinline__ float waveMax(float v) {
  #pragma unroll
  for (int off = 16; off >= 1; off >>= 1) v = fmaxf(v, __shfl_xor(v, off, 32));
  return v;
}

// ---------------------------------------------------------------------------
// Edge pass A: ex[e,h] = exp(att[h] . leaky_relu(xl[src] + xr[dst]));
//              denom[dst,h] += ex.   One block (64*H threads) per edge.
// ---------------------------------------------------------------------------
template<int H>
__global__ __launch_bounds__(64 * H)
void edge_attn(const float* __restrict__ xl, const float* __restrict__ xr,
               const int* __restrict__ src, const int* __restrict__ dst,
               const float* __restrict__ att, float* __restrict__ ex,
               float* __restrict__ denom, int E0)
{
  const int e = blockIdx.x;
  int s, d;
  if (e < E0) { s = src[e]; d = dst[e]; } else { s = e - E0; d = s; }
  const int t  = threadIdx.x;           // t = h*64 + c
  const int HC = 64 * H;
  float v = xl[(size_t)s * HC + t] + xr[(size_t)d * HC + t];
  v = v > 0.0f ? v : NEG_SLOPE * v;
  float p = waveSum(v * att[t]);        // per-wave partial

  __shared__ float part[2 * H];
  const int wave = t >> 5;              // 2 waves per head
  if ((t & 31) == 0) part[wave] = p;
  __syncthreads();
  if (t < H) {
    const float logit = part[2 * t] + part[2 * t + 1];
    const float exv = expf(logit);
    ex[(size_t)e * H + t] = exv;
    unsafeAtomicAdd(&denom[(size_t)d * H + t], exv);
  }
}

// ---------------------------------------------------------------------------
// Edge pass B: agg[dst, :] += (ex[e,h] / (denom[dst,h]+1e-16)) * xl[src, :]
// ---------------------------------------------------------------------------
template<int H>
__global__ __launch_bounds__(64 * H)
void edge_aggr(const float* __restrict__ xl, const float* __restrict__ ex,
               const float* __restrict__ denom,
               const int* __restrict__ src, const int* __restrict__ dst,
               float* __restrict__ agg, int E0)
{
  const int e = blockIdx.x;
  int s, d;
  if (e < E0) { s = src[e]; d = dst[e]; } else { s = e - E0; d = s; }
  const int t  = threadIdx.x;
  const int h  = t >> 6;
  const int HC = 64 * H;
  const float alpha = ex[(size_t)e * H + h] /
                      (denom[(size_t)d * H + h] + 1e-16f);
  unsafeAtomicAdd(&agg[(size_t)d * HC + t], alpha * xl[(size_t)s * HC + t]);
}

// h[i] = relu(h[i] + bias[i & 127])   (layer-1 finalize, HC = 128)
__global__ void bias_relu128(float* __restrict__ h, const float* __restrict__ bias,
                             int total)
{
  const int i = blockIdx.x * blockDim.x + threadIdx.x;
  if (i < total) {
    const float v = h[i] + bias[i & 127];
    h[i] = v > 0.0f ? v : 0.0f;
  }
}

// Per node: h = log_softmax(h + b2) over 64 channels. One 64-thread block/node.
__global__ __launch_bounds__(64)
void logsoftmax_bias(float* __restrict__ h, const float* __restrict__ bias)
{
  const int node = blockIdx.x;
  const int c = threadIdx.x;            // 0..63
  float* row = h + (size_t)node * 64;
  const float v = row[c] + bias[c];

  __shared__ float pm[2], ps[2];
  const int wave = c >> 5;
  float m = waveMax(v);
  if ((c & 31) == 0) pm[wave] = m;
  __syncthreads();
  const float mx = fmaxf(pm[0], pm[1]);
  float sv = waveSum(expf(v - mx));
  if ((c & 31) == 0) ps[wave] = sv;
  __syncthreads();
  const float total = ps[0] + ps[1];
  row[c] = v - mx - logf(total);
}

// ---------------------------------------------------------------------------
extern "C" void kernel_launch(void* const* d_in, const int* in_sizes, int n_in,
                              void* d_out, int out_size, void* d_ws, size_t ws_size,
                              hipStream_t stream)
{
  const float* x    = (const float*)d_in[0];
  const int*   ei   = (const int*)  d_in[1];
  const float* Wl1  = (const float*)d_in[2];
  const float* Wr1  = (const float*)d_in[3];
  const float* att1 = (const float*)d_in[4];
  const float* b1   = (const float*)d_in[5];
  const float* Wl2  = (const float*)d_in[6];
  const float* Wr2  = (const float*)d_in[7];
  const float* att2 = (const float*)d_in[8];
  const float* b2   = (const float*)d_in[9];
  const float* We1  = (const float*)d_in[10];
  const float* be1  = (const float*)d_in[11];
  const float* We2  = (const float*)d_in[12];
  const float* be2  = (const float*)d_in[13];

  const int N    = in_sizes[0] / 128;   // 50000
  const int E0   = in_sizes[1] / 2;     // 800000
  const int Etot = E0 + N;              // + self loops
  const int* srcI = ei;
  const int* dstI = ei + E0;

  float* ws    = (float*)d_ws;
  float* bufA  = ws;                              // N*128
  float* bufB  = bufA + (size_t)N * 128;          // N*128
  float* bufC  = bufB + (size_t)N * 128;          // N*128
  float* denom = bufC + (size_t)N * 128;          // 2*N
  float* ex    = denom + (size_t)2 * N;           // 2*(E+N)

  const dim3 blk32(32);
  const int rowT = (N + 15) / 16;                 // 3125

  // ---- Layer 1: projections (WMMA) ----
  gemm_wmma<0><<<dim3(8, rowT), blk32, 0, stream>>>(x, Wl1, nullptr, nullptr, bufA, N, 128, 128);
  gemm_wmma<0><<<dim3(8, rowT), blk32, 0, stream>>>(x, Wr1, nullptr, nullptr, bufB, N, 128, 128);

  hipMemsetAsync(denom, 0, (size_t)2 * N * sizeof(float), stream);
  hipMemsetAsync(bufC, 0, (size_t)N * 128 * sizeof(float), stream);

  edge_attn<2><<<Etot, 128, 0, stream>>>(bufA, bufB, srcI, dstI, att1, ex, denom, E0);
  edge_aggr<2><<<Etot, 128, 0, stream>>>(bufA, ex, denom, srcI, dstI, bufC, E0);

  bias_relu128<<<((N * 128) + 255) / 256, 256, 0, stream>>>(bufC, b1, N * 128);

  // ---- Layer 2: projections (WMMA), bufA reused ----
  float* xl2 = bufA;
  float* xr2 = bufA + (size_t)N * 64;
  gemm_wmma<0><<<dim3(4, rowT), blk32, 0, stream>>>(bufC, Wl2, nullptr, nullptr, xl2, N, 128, 64);
  gemm_wmma<0><<<dim3(4, rowT), blk32, 0, stream>>>(bufC, Wr2, nullptr, nullptr, xr2, N, 128, 64);

  hipMemsetAsync(denom, 0, (size_t)N * sizeof(float), stream);
  hipMemsetAsync(bufB, 0, (size_t)N * 64 * sizeof(float), stream);

  edge_attn<1><<<Etot, 64, 0, stream>>>(xl2, xr2, srcI, dstI, att2, ex, denom, E0);
  edge_aggr<1><<<Etot, 64, 0, stream>>>(xl2, ex, denom, srcI, dstI, bufB, E0);

  // ---- log_softmax (+b2) in place; bufB now holds hs ----
  logsoftmax_bias<<<N, 64, 0, stream>>>(bufB, b2);

  // ---- KnowledgeEnhancer: out = hs + sigmoid(relu(hs@We1+be1)@We2+be2) ----
  gemm_wmma<1><<<dim3(4, rowT), blk32, 0, stream>>>(bufB, We1, be1, nullptr, bufC, N, 64, 64);
  gemm_wmma<2><<<dim3(4, rowT), blk32, 0, stream>>>(bufC, We2, be2, bufB, (float*)d_out, N, 64, 64);
}